// BasicLSTM_19009525252104
// MI455X (gfx1250) — compile-verified
//
#include <hip/hip_runtime.h>
#include <stdint.h>

// Problem dims (from reference): x is [B, T] row-major, params is 12 scalars.
#define B_ROWS 32768
#define T_LEN  1024
#define BLK    256            // threads per block == rows per block
#define CHUNK  32             // timesteps per TDM tile (power-of-2 for pad_interval)
#define PAD_ICODE 4           // pad after 2^(code+1) = 32 dwords (one tile row)
#define PAD_ACODE 3           // pad amount = code+1 = 4 dwords
#define STRIDE 36             // CHUNK + 4 pad dwords -> 16B-aligned LDS rows
#define NCHUNK (T_LEN / CHUNK)

typedef unsigned int u32x4_t __attribute__((ext_vector_type(4)));
typedef int          i32x4_t __attribute__((ext_vector_type(4)));
typedef int          i32x8_t __attribute__((ext_vector_type(8)));

#if __has_builtin(__builtin_amdgcn_tensor_load_to_lds)
#define USE_TDM 1
#else
#define USE_TDM 0
#endif

// ---------- fast transcendentals (map to v_exp_f32 / v_rcp_f32 / v_tanh_f32) ----------
__device__ __forceinline__ float exp2_fast(float x) {
#if __has_builtin(__builtin_amdgcn_exp2f)
  return __builtin_amdgcn_exp2f(x);
#else
  return __builtin_exp2f(x);
#endif
}
__device__ __forceinline__ float rcp_fast(float x) {
#if __has_builtin(__builtin_amdgcn_rcpf)
  return __builtin_amdgcn_rcpf(x);
#else
  return 1.0f / x;
#endif
}
#define LOG2E_F 1.44269504088896340736f
__device__ __forceinline__ float sigm_fast(float a) {
  // 1 / (1 + exp(-a))
  return rcp_fast(1.0f + exp2_fast(-a * LOG2E_F));
}
__device__ __forceinline__ float tanh_fast(float a) {
#if __has_builtin(__builtin_amdgcn_tanhf)
  return __builtin_amdgcn_tanhf(a);
#else
  // 1 - 2/(exp(2a)+1)
  return 1.0f - 2.0f * rcp_fast(1.0f + exp2_fast(a * (2.0f * LOG2E_F)));
#endif
}

// ---------- TDM: stage a [CHUNK x BLK] tile of x into LDS with row padding ----------
#if USE_TDM
__device__ __forceinline__ void tdm_issue(const float* gsrc, unsigned lds_off, int remaining_t) {
  unsigned long long ga = (unsigned long long)(uintptr_t)gsrc;

  // D# group 0: count=1, lds_addr, global_addr[56:0], type=2
  u32x4_t g0;
  g0[0] = 1u;                                                   // count = 1 (valid)
  g0[1] = lds_off;                                              // LDS byte address
  g0[2] = (unsigned)ga;                                         // global_addr[31:0]
  g0[3] = (unsigned)((ga >> 32) & 0x01FFFFFFull) | (2u << 30);  // global_addr[56:32] | type=2

  // D# group 1:
  //  [17:16] data_size=2 (4B)  [20] pad_enable=1
  //  [24:22] pad_interval=PAD_ICODE (32 dwords)  [31:25] pad_amount=PAD_ACODE (4 dwords)
  //  -> LDS row stride = STRIDE floats
  i32x8_t g1;
  g1[0] = (2 << 16) | (1 << 20) | (PAD_ICODE << 22) | (PAD_ACODE << 25);
  g1[1] = (remaining_t & 0xFFFF) << 16;   // tensor_dim0[15:0] = remaining timesteps in row
  g1[2] = (BLK & 0xFFFF) << 16;           // tensor_dim0[31:16]=0 | tensor_dim1[15:0]=256 rows
  g1[3] = (CHUNK << 16);                  // tensor_dim1[31:16]=0 | tile_dim0 = 32
  g1[4] = BLK;                            // tile_dim1 = 256, tile_dim2 = 0
  g1[5] = T_LEN;                          // tensor_dim0_stride[31:0] = 1024 elements
  g1[6] = 0;                              // stride hi | tensor_dim1_stride lo (unused, 2D)
  g1[7] = 0;

  i32x4_t z4 = (i32x4_t)0;                // groups 2/3 unused (2-D tensor)
#if defined(__clang_major__) && (__clang_major__ >= 23)
  i32x8_t z8 = (i32x8_t)0;
  __builtin_amdgcn_tensor_load_to_lds(g0, g1, z4, z4, z8, 0);
#else
  __builtin_amdgcn_tensor_load_to_lds(g0, g1, z4, z4, 0);
#endif
}
#endif  // USE_TDM

// One LSTM recurrence step (pure VALU; fg/ig/og/pl are independent -> ILP)
#define LSTM_STEP(xt) do {                                        \
    float fg = sigm_fast(sm * w_fg0 + (xt) * w_fg1 + b_fg0);      \
    float pl = tanh_fast(sm * w_in0 + (xt) * w_in1 + b_in0);      \
    float ig = sigm_fast(sm * w_ig0 + (xt) * w_ig1 + b_ig0);      \
    float og = sigm_fast(sm * w_og0 + (xt) * w_og1 + b_og0);      \
    lm = lm * fg + pl * ig;                                       \
    sm = tanh_fast(lm) * og;                                      \
  } while (0)

__global__ __launch_bounds__(BLK) void lstm_scan_kernel(const float* __restrict__ x,
                                                        const float* __restrict__ params,
                                                        float* __restrict__ out) {
  // 12 scalar params -> scalar loads (uniform)
  const float w_fg0 = params[0],  w_fg1 = params[1],  b_fg0 = params[2];
  const float w_ig0 = params[3],  w_ig1 = params[4],  b_ig0 = params[5];
  const float w_in0 = params[6],  w_in1 = params[7],  b_in0 = params[8];
  const float w_og0 = params[9],  w_og1 = params[10], b_og0 = params[11];

  const int r   = threadIdx.x;
  const int row = blockIdx.x * BLK + r;

  float lm = 0.0f, sm = 0.0f;

#if USE_TDM
  __shared__ float xs[2][BLK * STRIDE];  // 72 KB, double-buffered tiles (CDNA5: 320KB/WG)
  const float* grow0 = x + (size_t)blockIdx.x * BLK * T_LEN;  // block's first row

  if (threadIdx.x < 32) {  // wave 0 drives the TDM pipeline
    tdm_issue(grow0, (unsigned)(uintptr_t)&xs[0][0], T_LEN);
  }

  for (int c = 0; c < NCHUNK; ++c) {
    if (threadIdx.x < 32) {
      if (c + 1 < NCHUNK) {
        tdm_issue(grow0 + (c + 1) * CHUNK,
                  (unsigned)(uintptr_t)&xs[(c + 1) & 1][0],
                  T_LEN - (c + 1) * CHUNK);
        __builtin_amdgcn_s_wait_tensorcnt(1);  // chunk c done, c+1 may fly
      } else {
        __builtin_amdgcn_s_wait_tensorcnt(0);  // last chunk done
      }
    }
    __syncthreads();  // tile xs[c&1] ready for all waves

    // Bulk-read this thread's 32 staged timesteps (ds_load_b128 x8, off the chain)
    const float* bp = &xs[c & 1][r * STRIDE];
    float xv[CHUNK];
#pragma unroll
    for (int k = 0; k < CHUNK / 4; ++k) {
      float4 v = *reinterpret_cast<const float4*>(bp + 4 * k);
      xv[4 * k + 0] = v.x; xv[4 * k + 1] = v.y;
      xv[4 * k + 2] = v.z; xv[4 * k + 3] = v.w;
    }
#pragma unroll
    for (int k = 0; k < CHUNK; ++k) {
      LSTM_STEP(xv[k]);
    }
    __syncthreads();  // done reading xs[c&1] before it is overwritten next iter
  }
#else
  // Fallback: direct per-thread float4 loads (L2 absorbs the lane-strided pattern)
  const float* xr = x + (size_t)row * T_LEN;
  for (int t = 0; t < T_LEN; t += 8) {
    float4 a = *reinterpret_cast<const float4*>(xr + t);
    float4 b = *reinterpret_cast<const float4*>(xr + t + 4);
    LSTM_STEP(a.x); LSTM_STEP(a.y); LSTM_STEP(a.z); LSTM_STEP(a.w);
    LSTM_STEP(b.x); LSTM_STEP(b.y); LSTM_STEP(b.z); LSTM_STEP(b.w);
  }
#endif

  out[row] = sm;
}

extern "C" void kernel_launch(void* const* d_in, const int* in_sizes, int n_in,
                              void* d_out, int out_size, void* d_ws, size_t ws_size,
                              hipStream_t stream) {
  const float* x      = (const float*)d_in[0];  // [32768, 1024] f32
  const float* params = (const float*)d_in[1];  // [12] f32
  float* out          = (float*)d_out;          // [32768] f32 (shape [B,1])

  dim3 grid(B_ROWS / BLK);  // 128 blocks x 256 threads = 1024 wave32s
  lstm_scan_kernel<<<grid, BLK, 0, stream>>>(x, params, out);
}